// ScheduleBattery_86260123174048
// MI455X (gfx1250) — compile-verified
//
#include <hip/hip_runtime.h>
#include <hip/hip_bf16.h>

namespace {

constexpr int   T      = 24;
constexpr int   NVAR   = 72;    // 3T
constexpr int   NCON   = 144;   // 6T
constexpr float EPS_   = 0.01f;
constexpr float EFF_   = 0.9f;
constexpr float IN_MAX_  = 0.5f;
constexpr float OUT_MAX_ = 1.0f;
constexpr float BCAP_  = 1.0f;
constexpr float LAM_   = 0.1f;
constexpr float SIGMA_ = 0.1f;
constexpr int   NIT    = 30;

constexpr int BPW = 16;   // batch problems per wave == WMMA M

// Padded LDS strides (floats) to avoid 64-bank conflicts on strided reads.
constexpr int SV = 76;    // rows of width 72
constexpr int SC = 148;   // rows of width 144
constexpr int SE = 25;    // rows of width 24
constexpr int SA = 76;    // A-matrix row stride

typedef __attribute__((ext_vector_type(2))) float v2f;
typedef __attribute__((ext_vector_type(8))) float v8f;

// Equality matrix A (24 x 72): row 0 = e_48; row r>=1:
//   EFF*e_{r-1} - e_{24+r-1} + e_{48+r-1} - e_{48+r}
__device__ __forceinline__ float Aval(int r, int k) {
  if (r == 0) return (k == 48) ? 1.0f : 0.0f;
  const int i = r - 1;
  if (k == i)      return EFF_;
  if (k == 24 + i) return -1.0f;
  if (k == 48 + i) return  1.0f;
  if (k == 49 + i) return -1.0f;
  return 0.0f;
}

__device__ __forceinline__ v8f wmma4(v2f a, v2f b, v8f c) {
  // D = A(16x4,f32) * B(4x16,f32) + C(16x16,f32)
  return __builtin_amdgcn_wmma_f32_16x16x4_f32(false, a, false, b,
                                               (short)0, c, false, false);
}

// Y(16x24, stride SE) = X(16x72, stride SV) @ A^T      (K = 72, 2 N-tiles)
__device__ __forceinline__ void gemm_xAT(const float* __restrict__ X,
                                         const float* __restrict__ Alds,
                                         float* __restrict__ Y, int lane) {
  const int m  = lane & 15;   // row (M) for A-operand, col (N) for B-operand
  const int kh = lane >> 4;   // K half: lanes 16-31 hold K=2,3 of each chunk
  #pragma unroll
  for (int nt = 0; nt < 2; ++nt) {
    const int  n   = nt * 16 + m;
    const bool nok = (n < 24);
    v8f acc = {};
    #pragma unroll
    for (int kk = 0; kk < 18; ++kk) {
      const int k0 = kk * 4 + 2 * kh;
      v2f a, b;
      a.x = X[m * SV + k0];
      a.y = X[m * SV + k0 + 1];
      b.x = nok ? Alds[n * SA + k0]     : 0.0f;   // (A^T)[k][n] = A[n][k]
      b.y = nok ? Alds[n * SA + k0 + 1] : 0.0f;
      acc = wmma4(a, b, acc);
    }
    #pragma unroll
    for (int j = 0; j < 8; ++j)
      if (nok) Y[(j + 8 * kh) * SE + n] = acc[j];   // row M = j + 8*kh
  }
}

// Y(16x72, stride SV) = X(16x24, stride SE) @ A       (K = 24, 5 N-tiles)
__device__ __forceinline__ void gemm_xA(const float* __restrict__ X,
                                        const float* __restrict__ Alds,
                                        float* __restrict__ Y, int lane) {
  const int m  = lane & 15;
  const int kh = lane >> 4;
  #pragma unroll
  for (int nt = 0; nt < 5; ++nt) {
    const int  n   = nt * 16 + m;
    const bool nok = (n < 72);
    v8f acc = {};
    #pragma unroll
    for (int kk = 0; kk < 6; ++kk) {
      const int k0 = kk * 4 + 2 * kh;
      v2f a, b;
      a.x = X[m * SE + k0];
      a.y = X[m * SE + k0 + 1];
      b.x = nok ? Alds[k0 * SA + n]       : 0.0f;
      b.y = nok ? Alds[(k0 + 1) * SA + n] : 0.0f;
      acc = wmma4(a, b, acc);
    }
    #pragma unroll
    for (int j = 0; j < 8; ++j)
      if (nok) Y[(j + 8 * kh) * SV + n] = acc[j];
  }
}

} // namespace

__global__ __launch_bounds__(32)
void ScheduleBattery_ipm_kernel(const float* __restrict__ logp,
                                float* __restrict__ out, int nb) {
  __shared__ float Alds[24 * SA];
  __shared__ float z  [BPW * SV];
  __shared__ float cv [BPW * SV];
  __shared__ float wv [BPW * SV];   // diag(H)^-1
  __shared__ float r1 [BPW * SV];   // KKT top rhs; reused for step ratios
  __shared__ float dzb[BPW * SV];   // w.*r1, then dz
  __shared__ float va [BPW * SV];   // nu@A, then dnu@A
  __shared__ float s  [BPW * SC];
  __shared__ float lmb[BPW * SC];
  __shared__ float nub[BPW * SE];
  __shared__ float rp [BPW * SE];
  __shared__ float gA [BPW * SE];
  __shared__ float dnu[BPW * SE];
  __shared__ float mu_l[BPW];
  __shared__ float al_l[BPW];

  const int lane = threadIdx.x;
  const int b0   = blockIdx.x * BPW;

  __builtin_prefetch(&logp[(size_t)b0 * T], 0, 0);   // global_prefetch_b8

  // Constant A into LDS (shared B-operand for all WMMAs).
  for (int idx = lane; idx < 24 * 72; idx += 32) {
    const int r = idx / 72, k = idx % 72;
    Alds[r * SA + k] = Aval(r, k);
  }
  // c = [exp(lp), -exp(lp), -LAM*BCAP]; z=0, nu=0, s=1, lm=1.
  for (int idx = lane; idx < BPW * T; idx += 32) {
    const int b = idx / T, t = idx % T, gb = b0 + b;
    const float lp = (gb < nb) ? logp[(size_t)gb * T + t] : 0.0f;
    const float p  = __expf(lp);
    cv[b * SV + t]      = p;
    cv[b * SV + 24 + t] = -p;
    cv[b * SV + 48 + t] = -LAM_ * BCAP_;
    nub[b * SE + t] = 0.0f;
  }
  for (int idx = lane; idx < BPW * NVAR; idx += 32)
    z[(idx / NVAR) * SV + (idx % NVAR)] = 0.0f;
  for (int idx = lane; idx < BPW * NCON; idx += 32) {
    const int a = (idx / NCON) * SC + (idx % NCON);
    s[a] = 1.0f; lmb[a] = 1.0f;
  }
  __syncthreads();

  #pragma unroll 1
  for (int it = 0; it < NIT; ++it) {
    // ---- mu[b] = mean(s .* lm): 2 lanes per batch, combine via shuffle.
    {
      const int b = lane >> 1, h = lane & 1;
      float acc = 0.0f;
      for (int j = h * 72; j < h * 72 + 72; ++j)
        acc += s[b * SC + j] * lmb[b * SC + j];
      acc += __shfl_xor(acc, 1, 32);
      if (h == 0) mu_l[b] = acc * (1.0f / 144.0f);
    }
    __syncthreads();

    gemm_xA(nub, Alds, va, lane);          // va = nu @ A   (A^T nu per var)
    __syncthreads();

    // ---- r1 = -(rd + ((lm*rpi - rc)/s) @ G); w = 1/diag(H); rw = w.*r1.
    for (int idx = lane; idx < BPW * NVAR; idx += 32) {
      const int b = idx / NVAR, v = idx % NVAR;
      const int blk = v / 24, t = v % 24;
      const int jp = b * SC + blk * 48 + t;        // +x_v <= h row
      const int jm = jp + 24;                      // -x_v <= 0 row
      const float Qd = (blk < 2) ? EPS_ : LAM_;
      const float hp = (blk == 0) ? IN_MAX_ : (blk == 1 ? OUT_MAX_ : BCAP_);
      const float zv = z[b * SV + v];
      const float sp = s[jp], sm = s[jm], lp = lmb[jp], lmm = lmb[jm];
      const float mu = mu_l[b];
      const float rd   = Qd * zv + cv[b * SV + v] + lp - lmm + va[b * SV + v];
      const float rpip =  zv + sp - hp;
      const float rpim = -zv + sm;
      const float rcp  = sp * lp  - SIGMA_ * mu;
      const float rcm  = sm * lmm - SIGMA_ * mu;
      const float up   = (lp  * rpip - rcp) / sp;
      const float um   = (lmm * rpim - rcm) / sm;
      const float r1v  = -(rd + up - um);
      const float wvv  = 1.0f / (Qd + lp / sp + lmm / sm);
      r1[b * SV + v]  = r1v;
      wv[b * SV + v]  = wvv;
      dzb[b * SV + v] = wvv * r1v;
    }
    __syncthreads();

    gemm_xAT(z,   Alds, rp, lane);         // rp = z @ A^T
    gemm_xAT(dzb, Alds, gA, lane);         // gA = (w.*r1) @ A^T
    __syncthreads();

    // ---- S = A W A^T is tridiagonal; solve S dnu = A W r1 + (A z - b)
    //      with Thomas, one lane per batch.
    if (lane < BPW) {
      const int b = lane;
      const float* wb = &wv[b * SV];
      float cp[24], gp[24];
      float e_prev = wb[48];                         // S[0][1] = +w3[0]
      {
        const float d0 = wb[48];
        const float g0 = gA[b * SE + 0] + rp[b * SE + 0] - 0.5f * BCAP_;
        cp[0] = e_prev / d0;
        gp[0] = g0 / d0;
      }
      #pragma unroll
      for (int r = 1; r < 24; ++r) {
        const float dr = EFF_ * EFF_ * wb[r - 1] + wb[24 + r - 1]
                       + wb[48 + r - 1] + wb[48 + r];
        const float er = -wb[48 + r];                // S[r][r+1]
        const float gr = gA[b * SE + r] + rp[b * SE + r];
        const float inv = 1.0f / (dr - e_prev * cp[r - 1]);
        cp[r] = er * inv;
        gp[r] = (gr - e_prev * gp[r - 1]) * inv;
        e_prev = er;
      }
      float x = gp[23];
      dnu[b * SE + 23] = x;
      #pragma unroll
      for (int r = 22; r >= 0; --r) {
        x = gp[r] - cp[r] * x;
        dnu[b * SE + r] = x;
      }
    }
    __syncthreads();

    gemm_xA(dnu, Alds, va, lane);          // va = dnu @ A  (A^T dnu per var)
    __syncthreads();

    // ---- dz = w.*(r1 - A^T dnu); per-element min step ratio into r1.
    for (int idx = lane; idx < BPW * NVAR; idx += 32) {
      const int b = idx / NVAR, v = idx % NVAR;
      const int blk = v / 24, t = v % 24;
      const int jp = b * SC + blk * 48 + t, jm = jp + 24;
      const float hp = (blk == 0) ? IN_MAX_ : (blk == 1 ? OUT_MAX_ : BCAP_);
      const float zv = z[b * SV + v];
      const float sp = s[jp], sm = s[jm], lp = lmb[jp], lmm = lmb[jm];
      const float mu = mu_l[b];
      const float dz = wv[b * SV + v] * (r1[b * SV + v] - va[b * SV + v]);
      const float rpip =  zv + sp - hp;
      const float rpim = -zv + sm;
      const float dsp = -rpip - dz;
      const float dsm = -rpim + dz;
      const float rcp = sp * lp  - SIGMA_ * mu;
      const float rcm = sm * lmm - SIGMA_ * mu;
      const float dlp  = (-rcp - lp  * dsp) / sp;
      const float dlmm = (-rcm - lmm * dsm) / sm;
      float r = 1e9f;
      if (dsp  < 0.0f) r = fminf(r, -sp  / dsp);
      if (dsm  < 0.0f) r = fminf(r, -sm  / dsm);
      if (dlp  < 0.0f) r = fminf(r, -lp  / dlp);
      if (dlmm < 0.0f) r = fminf(r, -lmm / dlmm);
      dzb[b * SV + v] = dz;
      r1[b * SV + v]  = r;      // reuse as ratio buffer
    }
    __syncthreads();

    // ---- alpha[b] = min(1, 0.99 * min_ratio[b])
    if (lane < BPW) {
      const int b = lane;
      float mn = 1e9f;
      for (int v = 0; v < NVAR; ++v) mn = fminf(mn, r1[b * SV + v]);
      al_l[b] = fminf(1.0f, 0.99f * mn);
    }
    __syncthreads();

    // ---- update z, s, lm (recompute ds/dlm from pre-update state), nu.
    for (int idx = lane; idx < BPW * NVAR; idx += 32) {
      const int b = idx / NVAR, v = idx % NVAR;
      const int blk = v / 24, t = v % 24;
      const int jp = b * SC + blk * 48 + t, jm = jp + 24;
      const float hp = (blk == 0) ? IN_MAX_ : (blk == 1 ? OUT_MAX_ : BCAP_);
      const float al = al_l[b];
      const float zv = z[b * SV + v];
      const float sp = s[jp], sm = s[jm], lp = lmb[jp], lmm = lmb[jm];
      const float mu = mu_l[b];
      const float dz = dzb[b * SV + v];
      const float rpip =  zv + sp - hp;
      const float rpim = -zv + sm;
      const float dsp = -rpip - dz;
      const float dsm = -rpim + dz;
      const float rcp = sp * lp  - SIGMA_ * mu;
      const float rcm = sm * lmm - SIGMA_ * mu;
      const float dlp  = (-rcp - lp  * dsp) / sp;
      const float dlmm = (-rcm - lmm * dsm) / sm;
      z[b * SV + v] = zv + al * dz;
      s[jp]   = sp  + al * dsp;
      s[jm]   = sm  + al * dsm;
      lmb[jp] = lp  + al * dlp;
      lmb[jm] = lmm + al * dlmm;
    }
    for (int idx = lane; idx < BPW * T; idx += 32) {
      const int b = idx / T, t = idx % T;
      nub[b * SE + t] += al_l[b] * dnu[b * SE + t];
    }
    __syncthreads();
  }

  // ---- write z (NB x 72, f32)
  for (int idx = lane; idx < BPW * NVAR; idx += 32) {
    const int b = idx / NVAR, v = idx % NVAR;
    const int gb = b0 + b;
    if (gb < nb) out[(size_t)gb * NVAR + v] = z[b * SV + v];
  }
}

extern "C" void kernel_launch(void* const* d_in, const int* in_sizes, int n_in,
                              void* d_out, int out_size, void* d_ws, size_t ws_size,
                              hipStream_t stream) {
  (void)n_in; (void)out_size; (void)d_ws; (void)ws_size;
  const float* logp = (const float*)d_in[0];
  float* out = (float*)d_out;
  const int nb = in_sizes[0] / T;                 // 8192 batch problems
  const int blocks = (nb + BPW - 1) / BPW;        // one wave32 per 16 problems
  ScheduleBattery_ipm_kernel<<<dim3(blocks), dim3(32), 0, stream>>>(logp, out, nb);
}